// GNNOnly_60249801228685
// MI455X (gfx1250) — compile-verified
//
#include <hip/hip_runtime.h>
#include <hip/hip_bf16.h>

// ---------------------------------------------------------------------------
// GNN (3x GraphConv + max-pool + MLP) for MI455X / gfx1250.
//
// Precision choice: fp32 end-to-end. The conv GEMMs are only ~10 GFLOP total
// while the edge gather/scatter moves ~2.5 GB (L2-resident: node features are
// 51.2 MB << 192 MB L2), so the workload is memory/atomic bound, not
// matrix-math bound. We therefore use the fp32-native WMMA
// V_WMMA_F32_16X16X4_F32 (full precision, still on the matrix pipes) instead
// of converting to bf16/fp8.
// ---------------------------------------------------------------------------

typedef float v2f __attribute__((ext_vector_type(2)));
typedef float v8f __attribute__((ext_vector_type(8)));

#define N_NODES_C 100000
#define N_EDGES_C 800000
#define HID_C     128
#define OUT_DIM_C 500
#define N_GRAPHS_C 64

// ---------------------------------------------------------------------------
// Degrees: one thread per edge, float atomic adds (L2 atomics).
// ---------------------------------------------------------------------------
__global__ void degree_kernel(const int* __restrict__ src, const int* __restrict__ dst,
                              float* __restrict__ odeg, float* __restrict__ ideg,
                              int n_edges) {
    int e = blockIdx.x * blockDim.x + threadIdx.x;
    if (e < n_edges) {
        atomicAdd(&odeg[src[e]], 1.0f);
        atomicAdd(&ideg[dst[e]], 1.0f);
    }
}

__global__ void norm_kernel(const float* __restrict__ odeg, const float* __restrict__ ideg,
                            float* __restrict__ srcn, float* __restrict__ dstn, int n) {
    int i = blockIdx.x * blockDim.x + threadIdx.x;
    if (i < n) {
        srcn[i] = rsqrtf(fmaxf(odeg[i], 1.0f));
        dstn[i] = rsqrtf(fmaxf(ideg[i], 1.0f));
    }
}

// ---------------------------------------------------------------------------
// Edge aggregation: one wave32 per edge; each lane moves a float4 (32*4=128).
// agg[dst] += h[src] * src_norm[src]   (global_atomic_add_f32, L2-resident)
// ---------------------------------------------------------------------------
__global__ void edge_agg_kernel(const float* __restrict__ h, const float* __restrict__ srcn,
                                const int* __restrict__ src, const int* __restrict__ dst,
                                float* __restrict__ agg, int n_edges) {
    int gid  = blockIdx.x * blockDim.x + threadIdx.x;
    int e    = gid >> 5;
    int lane = gid & 31;
    if (e >= n_edges) return;
    int s = src[e];
    int d = dst[e];
    float sn = srcn[s];
    float4 v = ((const float4*)(h + (size_t)s * HID_C))[lane];
    float* ap = agg + (size_t)d * HID_C + lane * 4;
    atomicAdd(ap + 0, v.x * sn);
    atomicAdd(ap + 1, v.y * sn);
    atomicAdd(ap + 2, v.z * sn);
    atomicAdd(ap + 3, v.w * sn);
}

// ---------------------------------------------------------------------------
// Conv GEMM: out[node, :] = relu( (agg[node,:] * dstn[node]) @ W + b )
// Block = 256 threads = 8 waves. Block owns a 16-row (M) tile; A tile
// (16x128, pre-scaled by dst_norm) staged in LDS with stride 132 so the 16
// rows a wave reads land in 16 distinct banks. Each wave owns one 16-column
// slice of W/out and runs 32x V_WMMA_F32_16X16X4_F32 over K=128.
//
// f32 WMMA fragment layout (ISA 7.12.2):
//   A 16x4 : lanes 0-15 -> {K=0,K=1}, lanes 16-31 -> {K=2,K=3}, M = lane%16
//   B 4x16 : lanes 0-15 -> {K=0,K=1}, lanes 16-31 -> {K=2,K=3}, N = lane%16
//   C/D    : v8f, element i at row M = i + (lane<16 ? 0 : 8), col = lane%16
// ---------------------------------------------------------------------------
__global__ void __launch_bounds__(256) conv_gemm_kernel(
    const float* __restrict__ agg, const float* __restrict__ dstn,
    const float* __restrict__ W, const float* __restrict__ bias,
    float* __restrict__ out, int n_nodes) {
    __shared__ float As[16][HID_C + 4];   // 132-float stride: conflict-free A reads

    const int tid   = threadIdx.x;
    const int mbase = blockIdx.x * 16;

    // Stage + scale the 16x128 A tile (512 float4 chunks / 256 threads).
    for (int i = tid; i < 16 * (HID_C / 4); i += 256) {
        int r    = i >> 5;      // row within tile (128/4 = 32 chunks per row)
        int c4   = i & 31;
        int node = mbase + r;
        float4 v = make_float4(0.f, 0.f, 0.f, 0.f);
        float  sc = 0.f;
        if (node < n_nodes) {
            v  = *(const float4*)(agg + (size_t)node * HID_C + c4 * 4);
            sc = dstn[node];
        }
        As[r][c4 * 4 + 0] = v.x * sc;
        As[r][c4 * 4 + 1] = v.y * sc;
        As[r][c4 * 4 + 2] = v.z * sc;
        As[r][c4 * 4 + 3] = v.w * sc;
    }
    __syncthreads();

    const int lane  = tid & 31;
    const int wave  = tid >> 5;
    const int l15   = lane & 15;
    const int khalf = (lane >> 4) << 1;     // 0 for lanes 0-15, 2 for lanes 16-31
    const int col   = wave * 16 + l15;      // this wave's output column

    v8f acc = {0.f, 0.f, 0.f, 0.f, 0.f, 0.f, 0.f, 0.f};
    #pragma unroll
    for (int k = 0; k < HID_C; k += 4) {
        v2f a, b;
        a.x = As[l15][k + khalf];
        a.y = As[l15][k + khalf + 1];
        b.x = W[(size_t)(k + khalf) * HID_C + col];
        b.y = W[(size_t)(k + khalf + 1) * HID_C + col];
        acc = __builtin_amdgcn_wmma_f32_16x16x4_f32(false, a, false, b,
                                                    (short)0, acc, false, false);
    }

    const float bv = bias[col];
    #pragma unroll
    for (int i = 0; i < 8; ++i) {
        int node = mbase + i + ((lane >> 4) << 3);
        if (node < n_nodes)
            out[(size_t)node * HID_C + col] = fmaxf(acc[i] + bv, 0.0f);
    }
}

// ---------------------------------------------------------------------------
// Generic bounds-checked WMMA GEMM (one wave per 16x16 output tile) for the
// tiny classifier matmuls. C = [relu](A[MxK] @ B[KxN] + bias). K % 4 == 0.
// ---------------------------------------------------------------------------
__global__ void __launch_bounds__(32) wmma_gemm_kernel(
    const float* __restrict__ A, const float* __restrict__ B,
    const float* __restrict__ bias, float* __restrict__ C,
    int M, int N, int K, int do_relu) {
    const int lane  = threadIdx.x & 31;
    const int l15   = lane & 15;
    const int khalf = (lane >> 4) << 1;
    const int mbase = blockIdx.x * 16;
    const int nbase = blockIdx.y * 16;
    const int row   = mbase + l15;
    const int col   = nbase + l15;

    v8f acc = {0.f, 0.f, 0.f, 0.f, 0.f, 0.f, 0.f, 0.f};
    for (int k = 0; k < K; k += 4) {
        int k0 = k + khalf, k1 = k + khalf + 1;
        v2f a, b;
        a.x = (row < M) ? A[(size_t)row * K + k0] : 0.0f;
        a.y = (row < M) ? A[(size_t)row * K + k1] : 0.0f;
        b.x = (col < N) ? B[(size_t)k0 * N + col] : 0.0f;
        b.y = (col < N) ? B[(size_t)k1 * N + col] : 0.0f;
        acc = __builtin_amdgcn_wmma_f32_16x16x4_f32(false, a, false, b,
                                                    (short)0, acc, false, false);
    }

    if (col < N) {
        const float bv = bias[col];
        #pragma unroll
        for (int i = 0; i < 8; ++i) {
            int m = mbase + i + ((lane >> 4) << 3);
            if (m < M) {
                float v = acc[i] + bv;
                C[(size_t)m * N + col] = do_relu ? fmaxf(v, 0.0f) : v;
            }
        }
    }
}

// ---------------------------------------------------------------------------
// Per-graph max pool. ReLU outputs are >= 0 and hg is zero-initialized, so a
// uint atomicMax on the float bit pattern is order-preserving.
// ---------------------------------------------------------------------------
__global__ void segmax_kernel(const float* __restrict__ h, const int* __restrict__ gids,
                              float* __restrict__ hg, int n_nodes) {
    int node = blockIdx.x;
    int c    = threadIdx.x;   // 128 cols
    if (node < n_nodes) {
        int g = gids[node];
        float v = h[(size_t)node * HID_C + c];
        atomicMax((unsigned int*)&hg[(size_t)g * HID_C + c], __float_as_uint(v));
    }
}

// ---------------------------------------------------------------------------
// Host launcher
// Inputs (setup_inputs order):
//  0 feat[N,128] 1 src[E] 2 dst[E] 3 graph_ids[N]
//  4 Wg1 5 bg1 6 Wg2 7 bg2 8 Wg3 9 bg3 10 Wc1 11 bc1 12 Wc2 13 bc2 14 Wc3 15 bc3
// Output: [64, 500] fp32.
// ---------------------------------------------------------------------------
extern "C" void kernel_launch(void* const* d_in, const int* in_sizes, int n_in,
                              void* d_out, int out_size, void* d_ws, size_t ws_size,
                              hipStream_t stream) {
    (void)in_sizes; (void)n_in; (void)out_size; (void)ws_size;

    const float* feat = (const float*)d_in[0];
    const int*   src  = (const int*)d_in[1];
    const int*   dst  = (const int*)d_in[2];
    const int*   gids = (const int*)d_in[3];
    const float* Wg[3] = {(const float*)d_in[4], (const float*)d_in[6], (const float*)d_in[8]};
    const float* bg[3] = {(const float*)d_in[5], (const float*)d_in[7], (const float*)d_in[9]};
    const float* Wc1 = (const float*)d_in[10]; const float* bc1 = (const float*)d_in[11];
    const float* Wc2 = (const float*)d_in[12]; const float* bc2 = (const float*)d_in[13];
    const float* Wc3 = (const float*)d_in[14]; const float* bc3 = (const float*)d_in[15];
    float* out = (float*)d_out;

    // Workspace carve-out (256B aligned slabs).
    char* ws = (char*)d_ws;
    size_t off = 0;
    auto take = [&](size_t elems) -> float* {
        float* p = (float*)(ws + off);
        off += ((elems * sizeof(float) + 255) / 256) * 256;
        return p;
    };
    float* odeg = take(N_NODES_C);
    float* ideg = take(N_NODES_C);
    float* srcn = take(N_NODES_C);
    float* dstn = take(N_NODES_C);
    float* agg  = take((size_t)N_NODES_C * HID_C);   // 51.2 MB
    float* h    = take((size_t)N_NODES_C * HID_C);   // 51.2 MB
    float* hg   = take((size_t)N_GRAPHS_C * HID_C);
    float* x1   = take((size_t)N_GRAPHS_C * HID_C);
    float* x2   = take((size_t)N_GRAPHS_C * (HID_C / 2));

    // Degrees + norms.
    hipMemsetAsync(odeg, 0, N_NODES_C * sizeof(float), stream);
    hipMemsetAsync(ideg, 0, N_NODES_C * sizeof(float), stream);
    degree_kernel<<<(N_EDGES_C + 255) / 256, 256, 0, stream>>>(src, dst, odeg, ideg, N_EDGES_C);
    norm_kernel<<<(N_NODES_C + 255) / 256, 256, 0, stream>>>(odeg, ideg, srcn, dstn, N_NODES_C);

    // 3x GraphConv: zero agg -> edge scatter-add -> WMMA GEMM (+bias, relu).
    const int mblocks = (N_NODES_C + 15) / 16;
    const float* hin = feat;
    for (int L = 0; L < 3; ++L) {
        hipMemsetAsync(agg, 0, (size_t)N_NODES_C * HID_C * sizeof(float), stream);
        edge_agg_kernel<<<(N_EDGES_C * 32 + 255) / 256, 256, 0, stream>>>(
            hin, srcn, src, dst, agg, N_EDGES_C);
        conv_gemm_kernel<<<mblocks, 256, 0, stream>>>(agg, dstn, Wg[L], bg[L], h, N_NODES_C);
        hin = h;
    }

    // Per-graph max pool.
    hipMemsetAsync(hg, 0, (size_t)N_GRAPHS_C * HID_C * sizeof(float), stream);
    segmax_kernel<<<N_NODES_C, HID_C, 0, stream>>>(h, gids, hg, N_NODES_C);

    // Classifier MLP (all WMMA).
    wmma_gemm_kernel<<<dim3(N_GRAPHS_C / 16, HID_C / 16), 32, 0, stream>>>(
        hg, Wc1, bc1, x1, N_GRAPHS_C, HID_C, HID_C, 1);
    wmma_gemm_kernel<<<dim3(N_GRAPHS_C / 16, (HID_C / 2) / 16), 32, 0, stream>>>(
        x1, Wc2, bc2, x2, N_GRAPHS_C, HID_C / 2, HID_C, 1);
    wmma_gemm_kernel<<<dim3(N_GRAPHS_C / 16, (OUT_DIM_C + 15) / 16), 32, 0, stream>>>(
        x2, Wc3, bc3, out, N_GRAPHS_C, OUT_DIM_C, HID_C / 2, 0);
}